// ROIPooling_42872363548706
// MI455X (gfx1250) — compile-verified
//
#include <hip/hip_runtime.h>
#include <cmath>

// ROI max-pool, MI455X (gfx1250).
// Feature map (1,512,40,40) f32 is fully L2-resident (3.2MB << 192MB L2), so the
// kernel is bound by the 51MB output write + L2 gather traffic. We use the CDNA5
// Tensor Data Mover to DMA each block's 3-D tile (CC channels x roi_h x roi_w)
// into LDS with a single tensor_load_to_lds, then max-reduce from LDS.

#define C_TOT    512
#define H_DIM    40
#define W_DIM    40
#define OUT_HW   7
#define NUM_ROIS 512
#define CC       8      // channels per block
#define MAX_S    36     // max ROI side in feature coords

typedef __attribute__((ext_vector_type(4))) unsigned int u32x4;
typedef __attribute__((ext_vector_type(8))) int          i32x8;
typedef __attribute__((ext_vector_type(4))) int          i32x4;

__global__ __launch_bounds__(256) void roi_pool_tdm_kernel(
    const float* __restrict__ fm,   // (1, 512, 40, 40)
    const float* __restrict__ roi,  // (512, 5): [b, x0, y0, x1, y1] * 16
    float* __restrict__ out)        // (512, 512, 7, 7)
{
    __shared__ float tile[CC * MAX_S * MAX_S];   // 41472 bytes

    const int r    = blockIdx.x;       // ROI index
    const int c0   = blockIdx.y * CC;  // first channel of this block
    const int tid  = threadIdx.x;

    // Decode ROI (uniform per block). ROI floats are exact ints / SCALE, so
    // round(x * 1/16) recovers the integer feature-map coords exactly.
    const float* rp = roi + (size_t)r * 5;
    const int x0 = (int)rintf(rp[1] * 0.0625f);
    const int y0 = (int)rintf(rp[2] * 0.0625f);
    const int x1 = (int)rintf(rp[3] * 0.0625f);
    const int y1 = (int)rintf(rp[4] * 0.0625f);
    const int rh = y1 - y0;            // roi height (rows), 8..36
    const int rw = x1 - x0;            // roi width  (cols), 8..36

    // ---- Phase 1: TDM load of tile [CC][rh][rw] into LDS (wave 0 issues) ----
    if (tid < 32) {
        // Low 32 bits of a generic pointer to LDS == LDS byte offset (ISA 10.2).
        const unsigned lds_base =
            (unsigned)(size_t)(void*)&tile[0];
        const unsigned long long ga =
            (unsigned long long)(size_t)(fm + ((size_t)c0 * H_DIM * W_DIM
                                               + (size_t)y0 * W_DIM + x0));

        // D# group 0 (128b): count=1 | lds_addr | global_addr[56:0] | type=2
        u32x4 g0;
        g0.x = 1u;                                      // count=1, user mode
        g0.y = lds_base;                                // lds_addr (bytes)
        g0.z = (unsigned)(ga & 0xFFFFFFFFu);            // global_addr[31:0]
        g0.w = (unsigned)((ga >> 32) & 0x1FFFFFFu)      // global_addr[56:32]
             | (2u << 30);                              // type = 2 ("image")

        // D# group 1 (256b): sizes/strides in data_size (4B) units.
        // dim0 = x (stride 1), dim1 = y (stride W), dim2 = channel (stride H*W)
        i32x8 g1;
        g1[0] = (2 << 16);                              // data_size = 4 bytes
        g1[1] = (rw & 0xFFFF) << 16;                    // tensor_dim0[15:0]
        g1[2] = (int)((((unsigned)rw) >> 16)            // tensor_dim0[31:16]
              | ((unsigned)(rh & 0xFFFF) << 16));       // tensor_dim1[15:0]
        g1[3] = (int)((((unsigned)rh) >> 16)            // tensor_dim1[31:16]
              | ((unsigned)rw << 16));                  // tile_dim0 = rw
        g1[4] = rh | (CC << 16);                        // tile_dim1, tile_dim2
        g1[5] = W_DIM;                                  // dim0_stride = 40
        g1[6] = (H_DIM * W_DIM) << 16;                  // dim1_stride[15:0]=1600
        g1[7] = 0;                                      // dim1_stride[47:16]

        // D# group 2: tensor_dim2 = CC (no dim3/iteration)
        i32x4 g2;
        g2[0] = CC;
        g2[1] = 0; g2[2] = 0; g2[3] = 0;
        i32x4 g3 = {0, 0, 0, 0};
        i32x8 g4 = {0, 0, 0, 0, 0, 0, 0, 0};            // extra group (unused)

        // amdgpu-toolchain (clang-23) 6-arg form; cpol = 0.
        __builtin_amdgcn_tensor_load_to_lds(g0, g1, g2, g3, g4, 0);
        __builtin_amdgcn_s_wait_tensorcnt(0);
    }
    __syncthreads();

    // ---- Phase 2: max-reduce 7x7 bins per channel out of LDS ----
    // Bug-faithful: row bins use bin_WIDTH, col bins use bin_HEIGHT.
    const float bw = (float)rw * (1.0f / OUT_HW);
    const float bh = (float)rh * (1.0f / OUT_HW);
    const float rhf = (float)rh;
    const float rwf = (float)rw;

    for (int idx = tid; idx < CC * OUT_HW * OUT_HW; idx += 256) {
        const int cl  = idx / (OUT_HW * OUT_HW);
        const int bin = idx - cl * (OUT_HW * OUT_HW);
        const int oy  = bin / OUT_HW;
        const int ox  = bin - oy * OUT_HW;

        const int rs = (int)floorf((float)oy * bw);                    // rel y0
        const int re = (int)fminf(ceilf((float)(oy + 1) * bw), rhf);
        const int cs = (int)floorf((float)ox * bh);                    // rel x0
        const int ce = (int)fminf(ceilf((float)(ox + 1) * bh), rwf);

        const float* base = &tile[cl * rh * rw];
        float m = -INFINITY;
        for (int y = rs; y < re; ++y) {
            const float* row = base + y * rw;
            for (int x = cs; x < ce; ++x)
                m = fmaxf(m, row[x]);
        }
        out[((size_t)r * C_TOT + (c0 + cl)) * (OUT_HW * OUT_HW) + bin] = m;
    }
}

extern "C" void kernel_launch(void* const* d_in, const int* in_sizes, int n_in,
                              void* d_out, int out_size, void* d_ws, size_t ws_size,
                              hipStream_t stream) {
    (void)in_sizes; (void)n_in; (void)out_size; (void)d_ws; (void)ws_size;
    const float* fm  = (const float*)d_in[0];   // (1,512,40,40) f32
    const float* roi = (const float*)d_in[1];   // (512,5) f32
    float* out = (float*)d_out;                 // (512,512,7,7) f32

    dim3 grid(NUM_ROIS, C_TOT / CC);
    roi_pool_tdm_kernel<<<grid, 256, 0, stream>>>(fm, roi, out);
}